// CTCLayer_6828998001144
// MI455X (gfx1250) — compile-verified
//
#include <hip/hip_runtime.h>

// CTC forward (batch cost) for B=1024, T=512, C=128, L=64  (S = 2L+1 = 129, blank = C-1)
// One extended-label state per lane, 2 batch rows per 256-thread block.
// Probability rows are streamed HBM -> LDS with CDNA5 async loads
// (global_load_async_to_lds_b128 / s_wait_asynccnt), triple-buffered so the
// T=512 sequential scan needs a single workgroup barrier per timestep.

#define B_    1024
#define T_    512
#define C_    128
#define L_    64
#define S_    129
#define BLANK 127
#define NEGV  (-1.0e9f)
#define EPSV  (1.0e-7f)
#define BPB   2          // batch rows per block

#if __has_builtin(__builtin_amdgcn_global_load_async_to_lds_b128)
#define HAVE_ASYNC_LDS 1
#else
#define HAVE_ASYNC_LDS 0
#endif

// Matches the builtin's parameter type: pointer to GCC vector of 4 ints.
typedef int v4i_ __attribute__((vector_size(4 * sizeof(int))));

__device__ __forceinline__ void issue_row_copy16(const float* gp, float* lds_dst) {
#if HAVE_ASYNC_LDS
  // 16 bytes per loader lane; 32 lanes cover one 512B probability row.
  __builtin_amdgcn_global_load_async_to_lds_b128(
      (__attribute__((address_space(1))) v4i_*)(const void*)gp,
      (__attribute__((address_space(3))) v4i_*)lds_dst,
      0 /*imm offset*/, 0 /*cpol*/);
#else
  const float4 v = *(const float4*)gp;
  *(float4*)lds_dst = v;
#endif
}

__device__ __forceinline__ void wait_async_le0() {
#if HAVE_ASYNC_LDS
#if __has_builtin(__builtin_amdgcn_s_wait_asynccnt)
  __builtin_amdgcn_s_wait_asynccnt(0);
#else
  asm volatile("s_wait_asynccnt 0x0" ::: "memory");
#endif
#endif
}
__device__ __forceinline__ void wait_async_le1() {
#if HAVE_ASYNC_LDS
#if __has_builtin(__builtin_amdgcn_s_wait_asynccnt)
  __builtin_amdgcn_s_wait_asynccnt(1);
#else
  asm volatile("s_wait_asynccnt 0x1" ::: "memory");
#endif
#endif
}
__device__ __forceinline__ void wait_async_le2() {
#if HAVE_ASYNC_LDS
#if __has_builtin(__builtin_amdgcn_s_wait_asynccnt)
  __builtin_amdgcn_s_wait_asynccnt(2);
#else
  asm volatile("s_wait_asynccnt 0x2" ::: "memory");
#endif
#endif
}

__global__ __launch_bounds__(128 * BPB)
void ctc_forward_kernel(const int* __restrict__ y_true,     // [B, L] int32
                        const float* __restrict__ y_pred,   // [B, T, C] softmax probs
                        float* __restrict__ out)            // [B, 1] -> flat [B]
{
  __shared__ float prob[BPB][3][C_];        // triple-buffered prob rows
  __shared__ float alpha[2][BPB][S_ + 3];   // ping-pong alpha (S=129, padded)

  const int  tid    = threadIdx.x;
  const int  half   = tid >> 7;             // which batch row within block
  const int  ls     = tid & 127;            // state id s for this lane (0..127)
  const int  b      = blockIdx.x * BPB + half;
  const bool loader = (ls < 32);            // wave 0 of each half issues DMA

  const float* rowbase = y_pred + (size_t)b * (size_t)(T_ * C_);

  // Extended label symbol for state s: odd s -> label[s/2], even s -> blank.
  const int lab = (ls & 1) ? y_true[b * L_ + (ls >> 1)] : BLANK;
  // diag-2 transition allowed: s odd, s>=3 (s==1 pads to NEG anyway),
  // and label differs from label at s-2.
  bool allow2 = false;
  if ((ls & 1) && ls >= 3)
    allow2 = (lab != y_true[b * L_ + (ls >> 1) - 1]);

  // Prime the pipeline: rows 0,1,2 in flight.
  if (loader) {
    const float* g0 = rowbase + ls * 4;
    issue_row_copy16(g0 + 0 * C_, &prob[half][0][ls * 4]);
    issue_row_copy16(g0 + 1 * C_, &prob[half][1][ls * 4]);
    issue_row_copy16(g0 + 2 * C_, &prob[half][2][ls * 4]);
    wait_async_le2();                       // row 0 complete (in-order)
  }
  __syncthreads();

  // t = 0 initialization: alpha0[0]=lp(0,blank), alpha0[1]=lp(0,label0), else NEG.
  {
    float a = NEGV;
    if (ls == 0) a = __logf(prob[half][0][BLANK] + EPSV);
    if (ls == 1) a = __logf(prob[half][0][lab]   + EPSV);
    alpha[0][half][ls] = a;
    if (ls == 127) alpha[0][half][128] = NEGV;
  }

  for (int t = 1; t < T_; ++t) {
    if (loader) {
      if (t < T_ - 1) wait_async_le1();     // row t complete, row t+1 may fly
      else            wait_async_le0();     // final row
    }
    __syncthreads();  // (a) row t visible to all waves
                      // (b) iter t-1 reads done -> safe to write alpha[t&1]
                      //     and to overwrite prob[(t+2)%3]

    if (loader && (t + 2 < T_))
      issue_row_copy16(rowbase + (size_t)(t + 2) * C_ + ls * 4,
                       &prob[half][(t + 2) % 3][ls * 4]);

    const float* pr = prob[half][t % 3];
    const float* aR = alpha[(t - 1) & 1][half];
    float*       aW = alpha[t & 1][half];

    // state s = ls
    {
      const int s  = ls;
      const float lp = __logf(pr[lab] + EPSV);
      const float a  = aR[s];
      const float a1 = (s >= 1) ? aR[s - 1] : NEGV;
      const float a2 = allow2 ? aR[s - 2] : NEGV;
      const float m  = fmaxf(a, fmaxf(a1, a2));
      const float sm = __expf(a - m) + __expf(a1 - m) + __expf(a2 - m);
      aW[s] = m + __logf(sm) + lp;
    }
    // extra state s = 128 (blank), owned by lane 127
    if (ls == 127) {
      const float lp = __logf(pr[BLANK] + EPSV);
      const float a  = aR[128];
      const float a1 = aR[127];
      const float m  = fmaxf(a, a1);
      const float sm = __expf(a - m) + __expf(a1 - m);
      aW[128] = m + __logf(sm) + lp;
    }
  }
  __syncthreads();

  if (ls == 0) {
    const float* aF = alpha[(T_ - 1) & 1][half];
    const float la = aF[S_ - 1];
    const float lb = aF[S_ - 2];
    const float m  = fmaxf(la, lb);
    const float ll = m + __logf(__expf(la - m) + __expf(lb - m));
    out[b] = -ll;
  }
}

extern "C" void kernel_launch(void* const* d_in, const int* in_sizes, int n_in,
                              void* d_out, int out_size, void* d_ws, size_t ws_size,
                              hipStream_t stream) {
  (void)in_sizes; (void)n_in; (void)out_size; (void)d_ws; (void)ws_size;
  const int*   y_true = (const int*)d_in[0];    // [B, L]
  const float* y_pred = (const float*)d_in[1];  // [B, T, C]
  float*       out    = (float*)d_out;          // [B, 1]
  dim3 grid(B_ / BPB);
  dim3 block(128 * BPB);
  hipLaunchKernelGGL(ctc_forward_kernel, grid, block, 0, stream, y_true, y_pred, out);
}